// MH2SD_7421703487608
// MI455X (gfx1250) — compile-verified
//
#include <hip/hip_runtime.h>
#include <hip/hip_bf16.h>
#include <math.h>

// ---------------------------------------------------------------------------
// MI455X fused kernel.
//
// Algebraic collapse of the reference:
//   v_spe1 = softmax over a size-1 axis  -> exactly 1.0 everywhere
//   out    = attn @ ones = row-sum of softmax -> exactly 1.0 everywhere
// => final = 1.0 + gelu(bn_inf(conv3x3_grouped(x4, local_w), bnc1_g, bnc1_b))
// where the reshape (b,128,24,24)->(b,576,128) is a flat reinterpretation,
// i.e. within a batch the output flat index is  o*576 + s  (o=out ch, s=i*24+j).
//
// Conv as implicit GEMM on V_WMMA_F32_16X16X4_F32 (fp32 in/out, no quant loss):
//   M = 16 spatial positions, N = 16 output channels, K = 64 in-ch x 9 taps.
// 1 block = 4 waves = the 4 N-tiles of one (batch, group, M-tile).
// Grid = B(2) * G(2) * 36 M-tiles = 144 blocks.
// ---------------------------------------------------------------------------

typedef __attribute__((ext_vector_type(2))) float v2f;
typedef __attribute__((ext_vector_type(8))) float v8f;

#define NP     24
#define NSP    (NP * NP)       /* 576 spatial positions   */
#define DIM    128
#define GC     64              /* channels per group      */
#define EPS_   1e-5f

__global__ __launch_bounds__(128) void fused_local_branch(
    const float* __restrict__ x,       // (B, N, DIM) flat == (B, DIM, 24, 24) flat
    const float* __restrict__ lw,      // local_w (128, 64, 3, 3)
    const float* __restrict__ gma,     // bnc1_gamma (128)
    const float* __restrict__ bta,     // bnc1_beta  (128)
    float* __restrict__ out)           // (B, N, 128)
{
    // Zero-padded input patch: 64 ch x 4 rows x 26 cols (halo rows/cols pre-zeroed)
    __shared__ float ldsA[GC][4][26];
    // Per-tap weight slab: per wave 64(K) x 16(N)
    __shared__ float ldsB[4][GC][16];

    const int tid  = threadIdx.x;
    const int wv   = tid >> 5;        // wave id 0..3  -> N-tile
    const int lane = tid & 31;
    const int hi   = lane >> 4;       // half-wave select
    const int nn   = lane & 15;       // N column / A row (M = lane%16)

    const int blk = blockIdx.x;       // 0..143
    const int mt  = blk % 36;         // M-tile
    const int bg  = blk / 36;
    const int b   = bg >> 1;
    const int g   = bg & 1;

    const int s0    = mt * 16;
    const int iA    = s0 / NP;        // first spatial row touched
    const int rbase = iA - 1;         // padded-row base

    const float* xb = x + b * (NSP * DIM);

    // ---- stage zero-padded input patch into LDS (branchless halo) ----------
    for (int e = tid; e < GC * 4 * 26; e += 128) {
        const int c   = e / 104;
        const int rem = e % 104;
        const int r   = rem / 26;
        const int jp  = rem % 26;
        const int row = rbase + r;
        const int col = jp - 1;
        float v = 0.0f;
        if ((unsigned)row < (unsigned)NP && (unsigned)col < (unsigned)NP)
            v = xb[(g * GC + c) * NSP + row * NP + col];
        ldsA[c][r][jp] = v;
    }

    // per-lane A-row coordinates (M = nn)
    const int sA = s0 + nn;
    const int iR = sA / NP;
    const int jR = sA % NP;
    const int r0 = iR - iA;           // 0 or 1

    const int obase = g * GC + wv * 16;   // first output channel of this wave

    v8f acc = {0.f, 0.f, 0.f, 0.f, 0.f, 0.f, 0.f, 0.f};

    for (int t = 0; t < 9; ++t) {
        __syncthreads();              // covers A-stage (t==0) and ldsB WAR (t>0)

        // stage this tap's 64x16 weight block for this wave
        for (int e = lane; e < GC * 16; e += 32) {
            const int k = e >> 4;
            const int n = e & 15;
            ldsB[wv][k][n] = lw[((obase + n) * GC + k) * 9 + t];
        }
        // prefetch next tap's weights into cache (global_prefetch_b8)
        if (t < 8)
            __builtin_prefetch(&lw[(obase + nn) * GC * 9 + (t + 1)], 0, 1);

        __syncthreads();

        const int ky   = t / 3;
        const int kx   = t - 3 * ky;
        const int ridx = r0 + ky;     // 0..3, halo rows are zero
        const int jp   = jR + kx;     // 0..25, halo cols are zero

        // K = 64 channels in 16 chunks of 4 -> 16 wmma per tap
        #pragma unroll
        for (int kc = 0; kc < 16; ++kc) {
            const int k0 = kc * 4 + 2 * hi;   // ISA: K = vgpr + 2*(lane>=16)
            v2f a, bm;
            a.x  = ldsA[k0    ][ridx][jp];
            a.y  = ldsA[k0 + 1][ridx][jp];
            bm.x = ldsB[wv][k0    ][nn];
            bm.y = ldsB[wv][k0 + 1][nn];
            acc = __builtin_amdgcn_wmma_f32_16x16x4_f32(
                      false, a, false, bm, (short)0, acc, false, false);
        }
    }

    // ---- epilogue: BN (inference) + exact GeLU + attention constant 1.0 ----
    const int   o  = obase + nn;
    const float sc = gma[o] * rsqrtf(1.0f + EPS_);
    const float bt = bta[o];

    // D layout: M = v + 8*hi, N = nn -> 8 consecutive spatial positions
    float* op = out + b * (NSP * DIM) + o * NSP + s0 + 8 * hi;
    #pragma unroll
    for (int v = 0; v < 8; ++v) {
        const float y = acc[v] * sc + bt;
        const float gel = 0.5f * y * (1.0f + erff(y * 0.70710678118654752f));
        op[v] = 1.0f + gel;
    }
}

extern "C" void kernel_launch(void* const* d_in, const int* in_sizes, int n_in,
                              void* d_out, int out_size, void* d_ws, size_t ws_size,
                              hipStream_t stream) {
    const float* x   = (const float*)d_in[0];    // x
    const float* lw  = (const float*)d_in[9];    // local_w
    const float* gma = (const float*)d_in[10];   // bnc1_gamma
    const float* bta = (const float*)d_in[11];   // bnc1_beta
    float* out = (float*)d_out;

    fused_local_branch<<<144, 128, 0, stream>>>(x, lw, gma, bta, out);
}